// TransformerGroupDecoder_8160437863112
// MI455X (gfx1250) — compile-verified
//
#include <hip/hip_runtime.h>
#include <math.h>

typedef __attribute__((ext_vector_type(2))) float v2f;
typedef __attribute__((ext_vector_type(8))) float v8f;

#define N_NODES   131072
#define D_FEAT    256
#define P_PROJ    64
#define E_EDGES   260096
#define G_GROUPS  8192
#define SIM_TH    0.5f

// ---------------------------------------------------------------------------
// Zero the group-sum accumulators (sums[G*D] followed by counts[G]).
// ---------------------------------------------------------------------------
__global__ void zero_ws_kernel(float* __restrict__ p, int n) {
    int i = blockIdx.x * blockDim.x + threadIdx.x;
    if (i < n) p[i] = 0.0f;
}

// ---------------------------------------------------------------------------
// BT[n][k] = B[k][n]  (64 KB, one-shot; makes the GEMM's B access contiguous
// per lane so each k-step's B fragment is a single global_load_b64).
// ---------------------------------------------------------------------------
__global__ void transpose_b_kernel(const float* __restrict__ B,   // [D_FEAT, P_PROJ]
                                   float* __restrict__ BT) {      // [P_PROJ, D_FEAT]
    int i = blockIdx.x * blockDim.x + threadIdx.x;  // < D_FEAT * P_PROJ
    int k = i >> 6;          // row of B (coalesced read)
    int n = i & (P_PROJ - 1);
    BT[(size_t)n * D_FEAT + k] = B[i];
}

// ---------------------------------------------------------------------------
// Fused: proj = features @ W (V_WMMA_F32_16X16X4_F32) + segment-sum scatter.
// Block = 128 threads = 4 waves; block owns 16 rows (M), wave w owns columns
// [16w,16w+16).
//
// Phase 1: branch-free pipelined WMMA loop (b64 A load + b64 BT load + wmma).
// Phase 2: wave w re-reads its 64-wide k-quarter of the A tile (4 KB, just
//          loaded -> WGP$ hit, no extra HBM traffic) and scatter-adds it into
//          the per-group sums. Straight-line, all lanes active, so phase 1
//          keeps clean s_clause batching and staged s_wait_loadcnt.
// ---------------------------------------------------------------------------
__global__ __launch_bounds__(128)
void gemm_scatter_kernel(const float* __restrict__ A,    // [N_NODES, D_FEAT]
                         const float* __restrict__ BT,   // [P_PROJ, D_FEAT]
                         const int*  __restrict__ gid,   // [N_NODES]
                         float* __restrict__ C,          // [N_NODES, P_PROJ]
                         float* __restrict__ sums,       // [G_GROUPS, D_FEAT]
                         float* __restrict__ counts) {   // [G_GROUPS]
    const int lane = threadIdx.x & 31;
    const int wave = threadIdx.x >> 5;
    const int m    = lane & 15;            // M (A) or N (B/C) within tile
    const int koff = (lane >> 4) << 1;     // lanes 16-31 hold K+2, K+3
    const int row0 = blockIdx.x << 4;      // 16 rows per block
    const int n0   = wave << 4;            // 16 cols per wave

    const float* arow  = A  + (size_t)(row0 + m) * D_FEAT + koff;  // A[m][k+koff..]
    const float* btrow = BT + (size_t)(n0 + m) * D_FEAT + koff;    // B[k+koff..][n]

    // ---- Phase 1: WMMA accumulation (no branches, EXEC all-ones) ----
    v8f acc = {};
    #pragma unroll 8
    for (int k = 0; k < D_FEAT; k += 4) {
        v2f a = *(const v2f*)(arow + k);    // 8B contiguous
        v2f b = *(const v2f*)(btrow + k);   // 8B contiguous (transposed W)
        acc = __builtin_amdgcn_wmma_f32_16x16x4_f32(
            /*neg_a=*/false, a, /*neg_b=*/false, b,
            /*c_mod=*/(short)0, acc, /*reuse_a=*/false, /*reuse_b=*/false);
    }

    // ---- Phase 2: segment-sum scatter of this wave's k-quarter ----
    const int g = gid[row0 + m];           // group of this lane's A row
    float* grow = sums + (size_t)g * D_FEAT + koff;
    const int kq_lo = wave << 6;           // [64w, 64w+64)
    #pragma unroll 4
    for (int k = kq_lo; k < kq_lo + 64; k += 4) {
        v2f a = *(const v2f*)(arow + k);   // WGP$/L0 hit (just streamed)
        atomicAdd(grow + k,     a.x);
        atomicAdd(grow + k + 1, a.y);
    }

    // Group size counts: one thread per block row.
    if (wave == 0 && lane < 16)
        atomicAdd(counts + g, 1.0f);

    // C/D layout: VGPR r -> row = r + 8*(lane>>4), col = lane&15
    const int mrow = row0 + ((lane >> 4) << 3);
    float* crow = C + (size_t)mrow * P_PROJ + n0 + (lane & 15);
    #pragma unroll
    for (int r = 0; r < 8; ++r)
        crow[(size_t)r * P_PROJ] = acc[r];
}

// ---------------------------------------------------------------------------
// One wave32 per edge: lanes cooperatively load both 64-float proj rows
// (coalesced 256B each), dot, shfl_xor tree reduce, sigmoid, write logit.
// ---------------------------------------------------------------------------
__global__ void edge_kernel(const float* __restrict__ proj,   // [N_NODES, P_PROJ]
                            const int*  __restrict__ eidx,    // [2, E_EDGES] flat
                            float* __restrict__ out) {        // [E_EDGES]
    int gtid = blockIdx.x * blockDim.x + threadIdx.x;
    int e    = gtid >> 5;
    int lane = gtid & 31;
    if (e >= E_EDGES) return;
    int s = eidx[e];
    int d = eidx[E_EDGES + e];
    v2f a = *(const v2f*)(proj + (size_t)s * P_PROJ + (lane << 1));
    v2f b = *(const v2f*)(proj + (size_t)d * P_PROJ + (lane << 1));
    float partial = a.x * b.x + a.y * b.y;
    #pragma unroll
    for (int off = 16; off > 0; off >>= 1)
        partial += __shfl_xor(partial, off, 32);
    if (lane == 0)
        out[e] = 1.0f / (1.0f + __expf(-(partial - SIM_TH)));
}

// ---------------------------------------------------------------------------
// means[i] = sums[i] / max(counts[i / D], 1)
// ---------------------------------------------------------------------------
__global__ void finalize_kernel(const float* __restrict__ sums,
                                const float* __restrict__ counts,
                                float* __restrict__ out) {     // [G_GROUPS*D_FEAT]
    int i = blockIdx.x * blockDim.x + threadIdx.x;
    float c = fmaxf(counts[i >> 8], 1.0f);   // D_FEAT == 256
    out[i] = sums[i] / c;
}

extern "C" void kernel_launch(void* const* d_in, const int* in_sizes, int n_in,
                              void* d_out, int out_size, void* d_ws, size_t ws_size,
                              hipStream_t stream) {
    const float* features   = (const float*)d_in[0];  // [N, D]
    const float* proj_W     = (const float*)d_in[1];  // [D, P]
    const int*   edge_index = (const int*)d_in[2];    // [2, E]
    const int*   group_ids  = (const int*)d_in[3];    // [N]
    // d_in[4] = num_groups (scalar) — fixed at G_GROUPS

    // Workspace: proj [N*P] | sums [G*D] | counts [G] | BT [P*D]
    float* proj   = (float*)d_ws;
    float* sums   = proj   + (size_t)N_NODES  * P_PROJ;
    float* counts = sums   + (size_t)G_GROUPS * D_FEAT;
    float* bt     = counts + (size_t)G_GROUPS;

    // Output layout: group_means [G*D] then edge_logits [E]
    float* out_means  = (float*)d_out;
    float* out_logits = out_means + (size_t)G_GROUPS * D_FEAT;

    const int zn = G_GROUPS * D_FEAT + G_GROUPS;
    zero_ws_kernel<<<(zn + 255) / 256, 256, 0, stream>>>(sums, zn);

    transpose_b_kernel<<<(D_FEAT * P_PROJ) / 256, 256, 0, stream>>>(proj_W, bt);

    gemm_scatter_kernel<<<N_NODES / 16, 128, 0, stream>>>(
        features, bt, group_ids, proj, sums, counts);

    edge_kernel<<<(E_EDGES * 32 + 255) / 256, 256, 0, stream>>>(
        proj, edge_index, out_logits);

    finalize_kernel<<<(G_GROUPS * D_FEAT) / 256, 256, 0, stream>>>(
        sums, counts, out_means);
}